// HeteroGAT_29807073034771
// MI455X (gfx1250) — compile-verified
//
#include <hip/hip_runtime.h>
#include <math.h>

#define HID 128
#define NH 4

typedef __attribute__((ext_vector_type(2))) float v2f;
typedef __attribute__((ext_vector_type(8))) float v8f;

static inline int ceil_div(int a, int b) { return (a + b - 1) / b; }

// ---------------------------------------------------------------------------
// sign-aware float atomic max (location must be initialized to -inf)
// ---------------------------------------------------------------------------
__device__ __forceinline__ void atomicMaxF(float* addr, float val) {
  if (val >= 0.0f) atomicMax((int*)addr, __float_as_int(val));
  else             atomicMin((unsigned int*)addr, __float_as_uint(val));
}

// ---------------------------------------------------------------------------
// fp32 WMMA GEMM: C[M,N] = A[M,K] @ W[K,N]  (+bias, optional ReLU)
// One wave computes a 16-row strip across all N (NT 16-col tiles).
// V_WMMA_F32_16X16X4_F32: A frag v2f (K split 0/1|2/3 by lane half),
// B frag v2f mirrored, C/D v8f (M = r + 8*laneHalf, N = lane&15).
// ---------------------------------------------------------------------------
template<int NT, bool SAFE>
__device__ __forceinline__ void gemm_kloop(const float* __restrict__ Arow,
                                           const float* __restrict__ W,
                                           int K, int ka_off, int l15, bool rok,
                                           v8f (&acc)[NT]) {
  constexpr int N = NT * 16;
  for (int k0 = 0; k0 < K; k0 += 4) {
    const int ka = k0 + ka_off;
    v2f a;
    if (SAFE) {
      a.x = rok ? Arow[ka]     : 0.0f;
      a.y = rok ? Arow[ka + 1] : 0.0f;
    } else {
      a = *(const v2f*)(Arow + ka);          // contiguous b64 load
    }
#pragma unroll
    for (int t = 0; t < NT; ++t) {
      const float* Wp = W + (size_t)ka * N + t * 16 + l15;
      v2f b;
      b.x = Wp[0];                           // W[ka  ][col]
      b.y = Wp[N];                           // W[ka+1][col]
      acc[t] = __builtin_amdgcn_wmma_f32_16x16x4_f32(
          false, a, false, b, (short)0, acc[t], false, false);
    }
  }
}

template<int NT>
__global__ __launch_bounds__(128) void wmma_gemm_f32(
    const float* __restrict__ A, const float* __restrict__ W,
    const float* __restrict__ bias, float* __restrict__ C,
    int M, int K, int relu)
{
  constexpr int N = NT * 16;
  const int wave = (int)((blockIdx.x * blockDim.x + threadIdx.x) >> 5);
  const int lane = (int)(threadIdx.x & 31);
  const int m0 = wave * 16;
  if (m0 >= M) return;                       // wave-uniform exit
  const int half = lane >> 4;
  const int l15  = lane & 15;
  const int rowA = m0 + l15;
  const int ka_off = half * 2;               // lanes 0-15: K=k0,k0+1 ; 16-31: k0+2,k0+3
  const float* Arow = A + (size_t)rowA * K;

  v8f acc[NT] = {};
  const bool full = (m0 + 16 <= M);          // wave-uniform

  if (full) {
    gemm_kloop<NT, false>(Arow, W, K, ka_off, l15, true, acc);
#pragma unroll
    for (int t = 0; t < NT; ++t) {
      const int col = t * 16 + l15;
      const float bv = bias ? bias[col] : 0.0f;
#pragma unroll
      for (int r = 0; r < 8; ++r) {
        const int row = m0 + r + half * 8;
        float v = acc[t][r] + bv;
        if (relu) v = v > 0.0f ? v : 0.0f;
        C[(size_t)row * N + col] = v;
      }
    }
  } else {
    gemm_kloop<NT, true>(Arow, W, K, ka_off, l15, rowA < M, acc);
#pragma unroll
    for (int t = 0; t < NT; ++t) {
      const int col = t * 16 + l15;
      const float bv = bias ? bias[col] : 0.0f;
#pragma unroll
      for (int r = 0; r < 8; ++r) {
        const int row = m0 + r + half * 8;
        if (row < M) {
          float v = acc[t][r] + bv;
          if (relu) v = v > 0.0f ? v : 0.0f;
          C[(size_t)row * N + col] = v;
        }
      }
    }
  }
}

static void launch_gemm(const float* A, const float* W, const float* bias,
                        float* C, int M, int K, int N, int relu, hipStream_t s) {
  int waves = (M + 15) / 16;
  dim3 grid(ceil_div(waves, 4)), block(128);
  if (N == 128) wmma_gemm_f32<8><<<grid, block, 0, s>>>(A, W, bias, C, M, K, relu);
  else          wmma_gemm_f32<4><<<grid, block, 0, s>>>(A, W, bias, C, M, K, relu);
}

// ---------------------------------------------------------------------------
// a[n,h] = sum_c h[n, h*32+c] * avec[h,c]
// ---------------------------------------------------------------------------
__global__ void att_score_k(const float* __restrict__ h, const float* __restrict__ avec,
                            float* __restrict__ out, int Nn) {
  int i = blockIdx.x * blockDim.x + threadIdx.x;
  if (i >= Nn * NH) return;
  int n = i >> 2, hh = i & 3;
  const float* hp = h + (size_t)n * HID + hh * 32;
  const float* ap = avec + hh * 32;
  float s = 0.0f;
#pragma unroll
  for (int c = 0; c < 32; ++c) s += hp[c] * ap[c];
  out[i] = s;
}

// m <- -inf, den <- 0, out <- broadcast(bias)
__global__ void conv_init_k(float* __restrict__ m, float* __restrict__ den,
                            float* __restrict__ out, const float* __restrict__ bias,
                            int Nd) {
  int i = blockIdx.x * blockDim.x + threadIdx.x;
  if (i < Nd * NH) { m[i] = -__builtin_inff(); den[i] = 0.0f; }
  if (i < Nd * HID) out[i] = bias[i & (HID - 1)];
}

__device__ __forceinline__ float lrelu(float v) { return v > 0.0f ? v : 0.2f * v; }

__global__ void edge_max_k(const int* __restrict__ src, const int* __restrict__ dst,
                           const float* __restrict__ as_, const float* __restrict__ ad_,
                           float* __restrict__ m, int E4) {
  int i = blockIdx.x * blockDim.x + threadIdx.x;
  if (i >= E4) return;
  int e = i >> 2, h = i & 3;
  int s = src[e], d = dst[e];
  atomicMaxF(&m[d * 4 + h], lrelu(as_[s * 4 + h] + ad_[d * 4 + h]));
}

__global__ void m_fix_k(float* __restrict__ m, int n) {
  int i = blockIdx.x * blockDim.x + threadIdx.x;
  if (i >= n) return;
  float v = m[i];
  if (!isfinite(v)) m[i] = 0.0f;
}

__global__ void edge_sum_k(const int* __restrict__ src, const int* __restrict__ dst,
                           const float* __restrict__ as_, const float* __restrict__ ad_,
                           const float* __restrict__ m, float* __restrict__ den, int E4) {
  int i = blockIdx.x * blockDim.x + threadIdx.x;
  if (i >= E4) return;
  int e = i >> 2, h = i & 3;
  int s = src[e], d = dst[e];
  float v = lrelu(as_[s * 4 + h] + ad_[d * 4 + h]);
  atomicAdd(&den[d * 4 + h], expf(v - m[d * 4 + h]));
}

// out[dst, c:c+4] += hs[src, c:c+4] * alpha(e, h)  -- 32 threads/edge, b128 gathers
__global__ void edge_scatter_k(const int* __restrict__ src, const int* __restrict__ dst,
                               const float* __restrict__ as_, const float* __restrict__ ad_,
                               const float* __restrict__ m, const float* __restrict__ den,
                               const float* __restrict__ hs, float* __restrict__ out, int E) {
  int i = blockIdx.x * blockDim.x + threadIdx.x;
  if (i >= E * 32) return;
  int e = i >> 5, q = i & 31;
  int c = q * 4, h = q >> 3;
  int s = src[e], d = dst[e];
  float v = lrelu(as_[s * 4 + h] + ad_[d * 4 + h]);
  float ex = expf(v - m[d * 4 + h]);
  float alpha = ex / (den[d * 4 + h] + 1e-16f);
  const float4 hv = *(const float4*)(hs + (size_t)s * HID + c);
  float* op = out + (size_t)d * HID + c;
  atomicAdd(op + 0, hv.x * alpha);
  atomicAdd(op + 1, hv.y * alpha);
  atomicAdd(op + 2, hv.z * alpha);
  atomicAdd(op + 3, hv.w * alpha);
}

// ---------------------------------------------------------------------------
// node-dimension softmax pooling:  x *= softmax(x @ attw + attb, axis=0)
// ---------------------------------------------------------------------------
__global__ void pool_init_k(float* sc) { sc[0] = -__builtin_inff(); sc[1] = 0.0f; }

__global__ void pool_z_k(const float* __restrict__ x, const float* __restrict__ attw,
                         const float* __restrict__ attb, float* __restrict__ z,
                         float* __restrict__ sc, int Nn) {
  int wave = (int)((blockIdx.x * blockDim.x + threadIdx.x) >> 5);
  int lane = (int)(threadIdx.x & 31);
  if (wave >= Nn) return;
  const float* xp = x + (size_t)wave * HID;
  float s = 0.0f;
#pragma unroll
  for (int i = 0; i < 4; ++i) { int c = lane + i * 32; s += xp[c] * attw[c]; }
  for (int off = 16; off; off >>= 1) s += __shfl_xor(s, off, 32);
  if (lane == 0) {
    s += attb[0];
    z[wave] = s;
    atomicMaxF(&sc[0], s);
  }
}

__global__ void pool_exp_k(float* __restrict__ z, float* __restrict__ sc, int Nn) {
  int n = blockIdx.x * blockDim.x + threadIdx.x;
  if (n >= Nn) return;
  float s = expf(z[n] - sc[0]);
  z[n] = s;
  atomicAdd(&sc[1], s);
}

__global__ void pool_scale_k(float* __restrict__ x, const float* __restrict__ z,
                             const float* __restrict__ sc, int Nn) {
  int i = blockIdx.x * blockDim.x + threadIdx.x;
  if (i >= Nn * HID) return;
  x[i] *= z[i >> 7] / sc[1];
}

// ---------------------------------------------------------------------------
// heads: pred = h_cls @ W2 + b2 ; risk = sigmoid(h_risk @ rW2 + rb2)
// ---------------------------------------------------------------------------
__global__ void heads_k(const float* __restrict__ hc, const float* __restrict__ hr,
                        const float* __restrict__ W2, const float* __restrict__ b2,
                        const float* __restrict__ rW2, const float* __restrict__ rb2,
                        float* __restrict__ pred, float* __restrict__ risk, int Nn) {
  int n = blockIdx.x * blockDim.x + threadIdx.x;
  if (n >= Nn) return;
  const float* p = hc + (size_t)n * 64;
  const float* q = hr + (size_t)n * 64;
  float p0 = b2[0], p1 = b2[1], r = rb2[0];
#pragma unroll 8
  for (int i = 0; i < 64; ++i) {
    float v = p[i];
    p0 += v * W2[i * 2];
    p1 += v * W2[i * 2 + 1];
    r  += q[i] * rW2[i];
  }
  pred[(size_t)n * 2]     = p0;
  pred[(size_t)n * 2 + 1] = p1;
  risk[n] = 1.0f / (1.0f + expf(-r));
}

// ---------------------------------------------------------------------------
// host orchestration
// ---------------------------------------------------------------------------
static void run_conv(const float* xs, int Ns, const float* xd, int Nd, int K,
                     const float* Wsrc, const float* Wdst, const float* asrc,
                     const float* adst, const float* bias,
                     const int* src, const int* dst, int E,
                     float* hs, float* hd, float* as_, float* ad_,
                     float* m, float* den, float* out, hipStream_t st) {
  launch_gemm(xs, Wsrc, nullptr, hs, Ns, K, 128, 0, st);
  launch_gemm(xd, Wdst, nullptr, hd, Nd, K, 128, 0, st);
  att_score_k  <<<ceil_div(Ns * 4, 256), 256, 0, st>>>(hs, asrc, as_, Ns);
  att_score_k  <<<ceil_div(Nd * 4, 256), 256, 0, st>>>(hd, adst, ad_, Nd);
  conv_init_k  <<<ceil_div(Nd * HID, 256), 256, 0, st>>>(m, den, out, bias, Nd);
  edge_max_k   <<<ceil_div(E * 4, 256), 256, 0, st>>>(src, dst, as_, ad_, m, E * 4);
  m_fix_k      <<<ceil_div(Nd * 4, 256), 256, 0, st>>>(m, Nd * 4);
  edge_sum_k   <<<ceil_div(E * 4, 256), 256, 0, st>>>(src, dst, as_, ad_, m, den, E * 4);
  edge_scatter_k<<<ceil_div(E * 32, 256), 256, 0, st>>>(src, dst, as_, ad_, m, den, hs, out, E);
}

static void run_pool(float* x, int Nn, const float* attw, const float* attb,
                     float* z, float* sc, hipStream_t st) {
  pool_init_k <<<1, 1, 0, st>>>(sc);
  pool_z_k    <<<ceil_div(Nn, 8), 256, 0, st>>>(x, attw, attb, z, sc, Nn);
  pool_exp_k  <<<ceil_div(Nn, 256), 256, 0, st>>>(z, sc, Nn);
  pool_scale_k<<<ceil_div(Nn * HID, 256), 256, 0, st>>>(x, z, sc, Nn);
}

extern "C" void kernel_launch(void* const* d_in, const int* in_sizes, int n_in,
                              void* d_out, int out_size, void* d_ws, size_t ws_size,
                              hipStream_t stream) {
  (void)n_in; (void)out_size; (void)ws_size;
  const float* x_card     = (const float*)d_in[0];
  const float* x_merchant = (const float*)d_in[1];
  const int NC = in_sizes[0] / 64;
  const int NM = in_sizes[1] / 64;
  const int E  = in_sizes[38];

  const float* W  [4][2]; const float* AV [4][2]; const float* B[4];
  for (int i = 0; i < 4; ++i) {               // 0:l0_cm 1:l0_mc 2:l1_cm 3:l1_mc
    int base = 2 + i * 5;
    W[i][0]  = (const float*)d_in[base + 0];  // Wsrc
    W[i][1]  = (const float*)d_in[base + 1];  // Wdst
    AV[i][0] = (const float*)d_in[base + 2];  // asrc
    AV[i][1] = (const float*)d_in[base + 3];  // adst
    B[i]     = (const float*)d_in[base + 4];  // bias
  }
  const float* attw_c = (const float*)d_in[22];
  const float* attb_c = (const float*)d_in[23];
  const float* attw_m = (const float*)d_in[24];
  const float* attb_m = (const float*)d_in[25];
  const float* cc_W1 = (const float*)d_in[26]; const float* cc_b1 = (const float*)d_in[27];
  const float* cc_W2 = (const float*)d_in[28]; const float* cc_b2 = (const float*)d_in[29];
  const float* cm_W1 = (const float*)d_in[30]; const float* cm_b1 = (const float*)d_in[31];
  const float* cm_W2 = (const float*)d_in[32]; const float* cm_b2 = (const float*)d_in[33];
  const float* rk_W1 = (const float*)d_in[34]; const float* rk_b1 = (const float*)d_in[35];
  const float* rk_W2 = (const float*)d_in[36]; const float* rk_b2 = (const float*)d_in[37];
  const int* cm_src = (const int*)d_in[38];
  const int* cm_dst = (const int*)d_in[39];
  const int* mc_src = (const int*)d_in[40];
  const int* mc_dst = (const int*)d_in[41];

  // workspace layout (floats)
  float* ws = (float*)d_ws;
  size_t off = 0;
  float* xc1  = ws + off; off += (size_t)NC * HID;
  float* xm1  = ws + off; off += (size_t)NM * HID;
  float* hs   = ws + off; off += (size_t)NC * HID;   // also reused as h_cls
  float* hd   = ws + off; off += (size_t)NC * HID;   // also reused as h_risk
  float* a_s  = ws + off; off += (size_t)NC * NH;
  float* a_d  = ws + off; off += (size_t)NC * NH;
  float* mbuf = ws + off; off += (size_t)NC * NH;
  float* den  = ws + off; off += (size_t)NC * NH;
  float* z    = ws + off; off += (size_t)NC;
  float* sc   = ws + off; off += 2;

  // output layout (floats, return order)
  float* out     = (float*)d_out;
  float* out_xc2 = out;
  float* out_xm2 = out_xc2 + (size_t)NC * HID;
  float* out_pc  = out_xm2 + (size_t)NM * HID;
  float* out_pm  = out_pc  + (size_t)NC * 2;
  float* out_rc  = out_pm  + (size_t)NM * 2;
  float* out_rm  = out_rc  + (size_t)NC;

  // layer 0
  run_conv(x_card, NC, x_merchant, NM, 64, W[0][0], W[0][1], AV[0][0], AV[0][1], B[0],
           cm_src, cm_dst, E, hs, hd, a_s, a_d, mbuf, den, xm1, stream);
  run_conv(x_merchant, NM, x_card, NC, 64, W[1][0], W[1][1], AV[1][0], AV[1][1], B[1],
           mc_src, mc_dst, E, hs, hd, a_s, a_d, mbuf, den, xc1, stream);
  // layer 1 (accumulate straight into d_out regions)
  run_conv(xc1, NC, xm1, NM, 128, W[2][0], W[2][1], AV[2][0], AV[2][1], B[2],
           cm_src, cm_dst, E, hs, hd, a_s, a_d, mbuf, den, out_xm2, stream);
  run_conv(xm1, NM, xc1, NC, 128, W[3][0], W[3][1], AV[3][0], AV[3][1], B[3],
           mc_src, mc_dst, E, hs, hd, a_s, a_d, mbuf, den, out_xc2, stream);

  // graph-level softmax pooling (in place on d_out)
  run_pool(out_xc2, NC, attw_c, attb_c, z, sc, stream);
  run_pool(out_xm2, NM, attw_m, attb_m, z, sc, stream);

  // classifier + risk heads (hs/hd reused as hidden activations)
  launch_gemm(out_xc2, cc_W1, cc_b1, hs, NC, 128, 64, 1, stream);
  launch_gemm(out_xc2, rk_W1, rk_b1, hd, NC, 128, 64, 1, stream);
  heads_k<<<ceil_div(NC, 256), 256, 0, stream>>>(hs, hd, cc_W2, cc_b2, rk_W2, rk_b2,
                                                 out_pc, out_rc, NC);
  launch_gemm(out_xm2, cm_W1, cm_b1, hs, NM, 128, 64, 1, stream);
  launch_gemm(out_xm2, rk_W1, rk_b1, hd, NM, 128, 64, 1, stream);
  heads_k<<<ceil_div(NM, 256), 256, 0, stream>>>(hs, hd, cm_W2, cm_b2, rk_W2, rk_b2,
                                                 out_pm, out_rm, NM);
}